// GCN_26499948216429
// MI455X (gfx1250) — compile-verified
//
#include <hip/hip_runtime.h>
#include <cmath>

typedef float v2f __attribute__((ext_vector_type(2)));
typedef float v8f __attribute__((ext_vector_type(8)));

__device__ __forceinline__ void atomicAddF32(float* p, float v) {
    // hardware global_atomic_add_f32, relaxed, device scope
    __hip_atomic_fetch_add(p, v, __ATOMIC_RELAXED, __HIP_MEMORY_SCOPE_AGENT);
}

// ---------------- degree / normalization ----------------
__global__ void k_fill1(float* __restrict__ p, int n) {
    int i = blockIdx.x * blockDim.x + threadIdx.x;
    if (i < n) p[i] = 1.0f;   // self-loop contributes 1 to deg(dst)
}

__global__ void k_deg(const int* __restrict__ dst, float* __restrict__ deg, int E) {
    int e = blockIdx.x * blockDim.x + threadIdx.x;
    if (e < E) atomicAddF32(&deg[dst[e]], 1.0f);
}

__global__ void k_rsqrt(float* __restrict__ p, int n) {
    int i = blockIdx.x * blockDim.x + threadIdx.x;
    if (i < n) p[i] = rsqrtf(p[i]);   // deg >= 1 always
}

// ---------------- WMMA GEMM: Y[n,OUT] = X[n,64] @ W[64,OUT] (+bias) ----------------
// One wave per 16-row tile; 4 waves (128 threads) per block. fp32 WMMA 16x16x4.
// W is staged in LDS *transposed* with stride 68 so each B fragment is one
// contiguous 8B ds_load_b64 and all 32 lanes hit distinct bank pairs.
template <int OUT>
__global__ void k_gemm_wmma(const float* __restrict__ X, const float* __restrict__ W,
                            const float* __restrict__ bias, float* __restrict__ Y, int n) {
    constexpr int WS = 68;                 // padded LDS row stride (floats)
    __shared__ __align__(16) float Wt[OUT * WS];
    const int tid = threadIdx.x;
    for (int i = tid; i < 64 * OUT; i += 128) {
        int k = i / OUT, c = i % OUT;      // W is row-major [64][OUT]
        Wt[c * WS + k] = W[i];
    }
    __syncthreads();

    const int lane  = tid & 31;
    const int tile  = blockIdx.x * 4 + (tid >> 5);
    const long tb   = (long)tile * 16;
    const int mrow  = lane & 15;           // A: matrix row within tile
    const int khalf = (lane >> 4) << 1;    // A/B: +0 or +2 in K per lane half
    const int ncol  = lane & 15;           // B/D: column within 16-col tile

    constexpr int CT = OUT / 16;
    v8f acc[CT];
#pragma unroll
    for (int c = 0; c < CT; ++c) acc[c] = (v8f){0.f,0.f,0.f,0.f,0.f,0.f,0.f,0.f};

    long arow = tb + mrow;
    if (arow >= n) arow = n - 1;           // clamp (EXEC must stay all-1 for WMMA)
    const float* __restrict__ xr = X + arow * 64 + khalf;
    const float* __restrict__ wr = Wt + khalf;   // + ncol-dependent row below

#pragma unroll 4
    for (int kt = 0; kt < 16; ++kt) {      // K = 64 in steps of 4
        v2f a = *(const v2f*)(xr + kt * 4);          // (K=k0, K=k0+1) for this lane
#pragma unroll
        for (int c = 0; c < CT; ++c) {
            v2f b = *(const v2f*)(wr + ((c << 4) + ncol) * WS + kt * 4);  // ds_load_b64
            acc[c] = __builtin_amdgcn_wmma_f32_16x16x4_f32(
                false, a, false, b, (short)0, acc[c], false, false);
        }
    }

    const int rhalf = (lane >> 4) << 3;    // D: lanes 16-31 hold rows M=8..15
#pragma unroll
    for (int c = 0; c < CT; ++c) {
#pragma unroll
        for (int i = 0; i < 8; ++i) {
            long r = tb + rhalf + i;
            if (r < n) {
                float v = acc[c][i];
                if (bias) v += bias[(c << 4) + ncol];
                Y[r * OUT + (c << 4) + ncol] = v;
            }
        }
    }
}

// ---------------- aggregation ----------------
// agg[i,:] = dinv[i]^2 * t[i,:]   (self-loop term)
__global__ void k_agg_init(const float* __restrict__ t, const float* __restrict__ dinv,
                           float* __restrict__ agg, int n) {
    long idx = (long)blockIdx.x * blockDim.x + threadIdx.x;
    if (idx < (long)n * 64) {
        int i = (int)(idx >> 6);
        float d = dinv[i];
        agg[idx] = d * d * t[idx];
    }
}

// one wave per edge: agg[dst,:] += dinv[src]*dinv[dst] * t[src,:]  (2 floats/lane)
__global__ void k_agg_edges(const int* __restrict__ src, const int* __restrict__ dst,
                            const float* __restrict__ dinv, const float* __restrict__ t,
                            float* __restrict__ agg, int E) {
    int e = blockIdx.x * (blockDim.x >> 5) + (threadIdx.x >> 5);
    if (e >= E) return;
    int lane = threadIdx.x & 31;
    int s = src[e], d = dst[e];
    float w = dinv[s] * dinv[d];
    v2f v = *(const v2f*)(t + (size_t)s * 64 + 2 * lane);   // L2-resident gather
    float* o = agg + (size_t)d * 64 + 2 * lane;
    atomicAddF32(o,     w * v.x);
    atomicAddF32(o + 1, w * v.y);
}

__global__ void k_tanh_bias(float* __restrict__ a, const float* __restrict__ b, int n) {
    long idx = (long)blockIdx.x * blockDim.x + threadIdx.x;
    if (idx < (long)n * 64) a[idx] = tanhf(a[idx] + b[idx & 63]);
}

// ---------------- host ----------------
extern "C" void kernel_launch(void* const* d_in, const int* in_sizes, int n_in,
                              void* d_out, int out_size, void* d_ws, size_t ws_size,
                              hipStream_t stream) {
    const float* x  = (const float*)d_in[0];
    const int*   ei = (const int*)d_in[1];
    const float* W1 = (const float*)d_in[2];
    const float* b1 = (const float*)d_in[3];
    const float* W2 = (const float*)d_in[4];
    const float* b2 = (const float*)d_in[5];
    const float* Wc = (const float*)d_in[6];
    const float* bc = (const float*)d_in[7];

    const int n = in_sizes[0] / 64;   // 100000
    const int E = in_sizes[1] / 2;    // 1200000
    const int* src = ei;
    const int* dst = ei + E;

    char* ws = (char*)d_ws;
    float* dinv = (float*)ws;
    size_t offT = (((size_t)n * 4) + 255) & ~(size_t)255;
    float* t   = (float*)(ws + offT);
    float* agg = (float*)(ws + offT + (size_t)n * 64 * 4);

    float* out = (float*)d_out;
    float* h2  = out + (size_t)n * 32;   // tuple: (out [n,32], h [n,64]) flat

    const int elemBlk  = (int)(((long)n * 64 + 255) / 256);
    const int nodeBlk  = (n + 255) / 256;
    const int edgeBlk  = (E + 255) / 256;
    const int edgeWBlk = (E + 7) / 8;                 // 8 waves/block, wave-per-edge
    const int tiles    = (n + 15) / 16;
    const int gemmBlk  = (tiles + 3) / 4;             // 4 waves/block

    // normalization: deg(dst) with self-loop, dinv = rsqrt(deg)
    k_fill1<<<nodeBlk, 256, 0, stream>>>(dinv, n);
    k_deg<<<edgeBlk, 256, 0, stream>>>(dst, dinv, E);
    k_rsqrt<<<nodeBlk, 256, 0, stream>>>(dinv, n);

    // layer 1: t = x@W1 ; h1 = tanh(D^-1/2 (A+I) D^-1/2 t + b1)  (h1 lives in agg)
    k_gemm_wmma<64><<<gemmBlk, 128, 0, stream>>>(x, W1, nullptr, t, n);
    k_agg_init<<<elemBlk, 256, 0, stream>>>(t, dinv, agg, n);
    k_agg_edges<<<edgeWBlk, 256, 0, stream>>>(src, dst, dinv, t, agg, E);
    k_tanh_bias<<<elemBlk, 256, 0, stream>>>(agg, b1, n);

    // layer 2: t = h1@W2 ; h2 = tanh(... + b2) aggregated directly into d_out h-region
    k_gemm_wmma<64><<<gemmBlk, 128, 0, stream>>>(agg, W2, nullptr, t, n);
    k_agg_init<<<elemBlk, 256, 0, stream>>>(t, dinv, h2, n);
    k_agg_edges<<<edgeWBlk, 256, 0, stream>>>(src, dst, dinv, t, h2, E);
    k_tanh_bias<<<elemBlk, 256, 0, stream>>>(h2, b2, n);

    // classifier: out = h2 @ Wc + bc
    k_gemm_wmma<32><<<gemmBlk, 128, 0, stream>>>(h2, Wc, bc, out, n);
}